// Wav2Vec2GumbelVectorQuantizer_1400159338917
// MI455X (gfx1250) — compile-verified
//
#include <hip/hip_runtime.h>
#include <hip/hip_bf16.h>

typedef __attribute__((ext_vector_type(2))) float v2f;
typedef __attribute__((ext_vector_type(4))) float v4f;
typedef __attribute__((ext_vector_type(8))) float v8f;

#define GQ     2
#define VQ     320
#define GV     640      // GQ*VQ
#define CDIM   512
#define DQ     128      // codevector dim per group
#define NROWS  32768    // B*L
#define NRG    65536    // B*L*GQ
#define TAUINV 0.5f

// ---------------------------------------------------------------------------
// Kernel 1: logits = hidden @ w_proj + b  via V_WMMA_F32_16X16X4_F32
// Block tile M=128 x N=64; 8 waves in 4(m) x 2(n); each wave: 2x2 16x16 tiles.
// A-frag layout (ISA 16x4 f32): lane<16 -> M=lane, K={k,k+1}; lane>=16 -> K={k+2,k+3}
// -> contiguous float2 per lane in row-major A. B-frag mirrors with N across lanes.
// ---------------------------------------------------------------------------
__global__ __launch_bounds__(256) void gemm_logits_kernel(
    const float* __restrict__ A,    // [32768, 512]
    const float* __restrict__ Bw,   // [512, 640]
    const float* __restrict__ bias, // [640]
    float* __restrict__ C)          // [32768, 640]
{
  const int tid   = threadIdx.x;
  const int wave  = tid >> 5;
  const int lane  = tid & 31;
  const int bm    = blockIdx.x / (GV / 64);
  const int bn    = blockIdx.x % (GV / 64);
  const int wm    = wave & 3;
  const int wn    = wave >> 2;
  const int m0    = bm * 128 + wm * 32;
  const int n0    = bn * 64  + wn * 32;
  const int lhalf = lane >> 4;   // selects K pair {0,1} vs {2,3}
  const int lidx  = lane & 15;

  v8f acc00 = {}; v8f acc01 = {}; v8f acc10 = {}; v8f acc11 = {};

  const float* Arow0 = A + (size_t)(m0 + lidx)      * CDIM + 2 * lhalf;
  const float* Arow1 = A + (size_t)(m0 + 16 + lidx) * CDIM + 2 * lhalf;

  for (int k = 0; k < CDIM; k += 4) {
    v2f a0 = *(const v2f*)(Arow0 + k);
    v2f a1 = *(const v2f*)(Arow1 + k);

    const float* Bp = Bw + (size_t)(k + 2 * lhalf) * GV + n0 + lidx;
    v2f b0, b1;
    b0.x = Bp[0];        b0.y = Bp[GV];        // N-subtile 0: K, K+1
    b1.x = Bp[16];       b1.y = Bp[GV + 16];   // N-subtile 1

    acc00 = __builtin_amdgcn_wmma_f32_16x16x4_f32(false, a0, false, b0, (short)0, acc00, false, false);
    acc01 = __builtin_amdgcn_wmma_f32_16x16x4_f32(false, a0, false, b1, (short)0, acc01, false, false);
    acc10 = __builtin_amdgcn_wmma_f32_16x16x4_f32(false, a1, false, b0, (short)0, acc10, false, false);
    acc11 = __builtin_amdgcn_wmma_f32_16x16x4_f32(false, a1, false, b1, (short)0, acc11, false, false);
  }

  // C/D 16x16 f32 layout: VGPR j: lane<16 -> M=j, lane>=16 -> M=8+j; N=lane&15
  const int colA = n0 + lidx;
  const int colB = n0 + 16 + lidx;
  const float bA = bias[colA];
  const float bB = bias[colB];
#pragma unroll
  for (int j = 0; j < 8; ++j) {
    int r0 = m0 + j + 8 * lhalf;
    int r1 = r0 + 16;
    C[(size_t)r0 * GV + colA] = acc00[j] + bA;
    C[(size_t)r0 * GV + colB] = acc01[j] + bB;
    C[(size_t)r1 * GV + colA] = acc10[j] + bA;
    C[(size_t)r1 * GV + colB] = acc11[j] + bB;
  }
}

// ---------------------------------------------------------------------------
// Kernel 2: per-row gumbel-softmax + argmax gather + clean-softmax partials.
// One wave (32 lanes) per row of 320; 8 rows per block. Deterministic.
// ---------------------------------------------------------------------------
__global__ __launch_bounds__(256) void gumbel_kernel(
    const float* __restrict__ logits,  // [65536, 320] (flat view)
    const float* __restrict__ gu,      // [65536, 320]
    const float* __restrict__ cv,      // [640, 128]
    float* __restrict__ out,           // [32768, 256]
    float* __restrict__ partial)       // [gridDim.x, 640]
{
  __shared__ float sacc[8][VQ];   // clean softmax per wave, no atomics

  const int wave = threadIdx.x >> 5;
  const int lane = threadIdx.x & 31;
  const int r    = blockIdx.x * 8 + wave;   // row in [0, 65536)
  const int g    = r & 1;                   // wave parity == group
  const int n    = r >> 1;

  const float* lrow = logits + (size_t)r * VQ;
  const float* urow = gu     + (size_t)r * VQ;

  float x[10], noisy[10];
  float nmax = -3.0e38f;  int nidx = 0;
  float cmax = -3.0e38f;
#pragma unroll
  for (int t = 0; t < 10; ++t) {
    int v = lane + t * 32;
    float xv  = lrow[v];
    float u   = urow[v];
    float gmb = -__logf(-__logf(u + 1e-10f) + 1e-10f);
    float nv  = (xv + gmb) * TAUINV;
    x[t] = xv; noisy[t] = nv;
    if (nv > nmax) { nmax = nv; nidx = v; }   // first (lowest v) wins ties
    cmax = fmaxf(cmax, xv);
  }
  // wave32 reductions: max+argmax (tie -> lower index), max of clean
#pragma unroll
  for (int off = 16; off > 0; off >>= 1) {
    float ov = __shfl_xor(nmax, off, 32);
    int   oi = __shfl_xor(nidx, off, 32);
    if (ov > nmax || (ov == nmax && oi < nidx)) { nmax = ov; nidx = oi; }
    cmax = fmaxf(cmax, __shfl_xor(cmax, off, 32));
  }

  float nsum = 0.0f, csum = 0.0f;
  float sc[10];
#pragma unroll
  for (int t = 0; t < 10; ++t) {
    nsum += __expf(noisy[t] - nmax);
    float e = __expf(x[t] - cmax);
    sc[t] = e;
    csum += e;
  }
#pragma unroll
  for (int off = 16; off > 0; off >>= 1) {
    nsum += __shfl_xor(nsum, off, 32);
    csum += __shfl_xor(csum, off, 32);
  }
  const float inv_csum = 1.0f / csum;
#pragma unroll
  for (int t = 0; t < 10; ++t)
    sacc[wave][lane + t * 32] = sc[t] * inv_csum;

  // straight-through: y_hard - sg(y_soft) + y_soft == ((1-p*)+p*) at argmax, 0 else
  const float pstar = 1.0f / nsum;          // exp(0)/sum after max-subtract
  const float scale = (1.0f - pstar) + pstar;
  const float* cvrow = cv + (size_t)(g * VQ + nidx) * DQ;
  float* orow = out + (size_t)n * (GQ * DQ) + g * DQ;
  v4f val = *(const v4f*)(cvrow + lane * 4);
  val.x *= scale; val.y *= scale; val.z *= scale; val.w *= scale;
  *(v4f*)(orow + lane * 4) = val;

  __syncthreads();
  // waves 0,2,4,6 -> g=0 ; 1,3,5,7 -> g=1 (blockIdx*8 is even)
  for (int i = threadIdx.x; i < GV; i += 256) {
    int gg = i / VQ, v = i % VQ;
    float s = sacc[gg][v] + sacc[gg + 2][v] + sacc[gg + 4][v] + sacc[gg + 6][v];
    partial[(size_t)blockIdx.x * GV + i] = s;
  }
}

// ---------------------------------------------------------------------------
// Kernel 3: column-sum partials [8192,640] -> colsum[640] (one block per col)
// ---------------------------------------------------------------------------
__global__ __launch_bounds__(256) void reduce_cols_kernel(
    const float* __restrict__ partial, float* __restrict__ colsum, int nblk)
{
  const int col = blockIdx.x;
  float s = 0.0f;
  for (int b = threadIdx.x; b < nblk; b += 256)
    s += partial[(size_t)b * GV + col];
  __shared__ float red[256];
  red[threadIdx.x] = s;
  __syncthreads();
  for (int off = 128; off > 0; off >>= 1) {
    if (threadIdx.x < off) red[threadIdx.x] += red[threadIdx.x + off];
    __syncthreads();
  }
  if (threadIdx.x == 0) colsum[col] = red[0];
}

// ---------------------------------------------------------------------------
// Kernel 4: perplexity = sum_g exp(-sum_v p log(p+1e-7)), p = colsum/32768
// ---------------------------------------------------------------------------
__global__ void perplexity_kernel(const float* __restrict__ colsum,
                                  float* __restrict__ outp)
{
  const int g = threadIdx.x >> 5;   // 2 waves, one per group
  const int lane = threadIdx.x & 31;
  float h = 0.0f;
  for (int v = lane; v < VQ; v += 32) {
    float p = colsum[g * VQ + v] * (1.0f / 32768.0f);
    h += p * __logf(p + 1e-7f);
  }
#pragma unroll
  for (int off = 16; off > 0; off >>= 1) h += __shfl_xor(h, off, 32);
  __shared__ float hg[2];
  if (lane == 0) hg[g] = h;
  __syncthreads();
  if (threadIdx.x == 0) outp[0] = __expf(-hg[0]) + __expf(-hg[1]);
}

// ---------------------------------------------------------------------------
extern "C" void kernel_launch(void* const* d_in, const int* in_sizes, int n_in,
                              void* d_out, int out_size, void* d_ws, size_t ws_size,
                              hipStream_t stream) {
  const float* hs = (const float*)d_in[0];  // [8,4096,512]
  const float* gu = (const float*)d_in[1];  // [65536,320]
  const float* w  = (const float*)d_in[2];  // [512,640]
  const float* b  = (const float*)d_in[3];  // [640]
  const float* cv = (const float*)d_in[4];  // [1,640,128]

  float* out     = (float*)d_out;                    // [32768,256] + 1 scalar
  float* logits  = (float*)d_ws;                     // 32768*640
  float* partial = logits + (size_t)NROWS * GV;      // 8192*640
  float* colsum  = partial + (size_t)(NRG / 8) * GV; // 640

  gemm_logits_kernel<<<dim3((NROWS / 128) * (GV / 64)), dim3(256), 0, stream>>>(
      hs, w, b, logits);
  gumbel_kernel<<<dim3(NRG / 8), dim3(256), 0, stream>>>(
      logits, gu, cv, out, partial);
  reduce_cols_kernel<<<dim3(GV), dim3(256), 0, stream>>>(partial, colsum, NRG / 8);
  perplexity_kernel<<<dim3(1), dim3(64), 0, stream>>>(
      colsum, out + (size_t)NROWS * GQ * DQ);
}